// GeneEmbeddingModel_78219944395132
// MI455X (gfx1250) — compile-verified
//
#include <hip/hip_runtime.h>
#include <hip/hip_bf16.h>

// ---------------------------------------------------------------------------
// GeneEmbeddingModel for MI455X (gfx1250, wave32)
//   - kNN graph build: distance Gram matrix via fp32 WMMA + chunked top-15
//     merge with half-wave shuffle reductions
//   - 2-layer MLP via V_WMMA_F32_16X16X4_F32 (fp32 WMMA, exact precision)
//   - 10-hop GPR-GNN propagation: CSC gather with async LDS double-buffering
//     (global_load_async_to_lds_b128 + s_wait_asynccnt)
//   - mean readout + prediction GEMM via fp32 WMMA
// ---------------------------------------------------------------------------

typedef __attribute__((ext_vector_type(2))) float v2f;
typedef __attribute__((ext_vector_type(8))) float v8f;

#define NN     8192   // nodes
#define BB     16     // batch (genes)
#define DD     128    // feature dim
#define HH     128    // hidden dim
#define KNEI   15     // neighbors
#define M_NEI  (NN * KNEI)     // neighbor edges
#define M_TOT  (NN * (KNEI+1)) // + self loops
#define BD     (BB * DD)       // 2048 floats per node
#define CHUNK  128             // distance columns per merge chunk

// ---------------------------------------------------------------------------
// init: deg = 1 (self loop weight 1), in_cnt = 1 (self edge), zero gene_rep
// ---------------------------------------------------------------------------
__global__ __launch_bounds__(256) void init_kernel(float* deg, int* in_cnt,
                                                   float* gene_rep) {
  int i = blockIdx.x * 256 + threadIdx.x;
  if (i < NN) { deg[i] = 1.0f; in_cnt[i] = 1; }
  if (i < BD) gene_rep[i] = 0.0f;
}

// ---------------------------------------------------------------------------
// squared norms of cell embeddings
// ---------------------------------------------------------------------------
__global__ __launch_bounds__(256) void sqnorm_kernel(const float* __restrict__ emb,
                                                     float* __restrict__ sq) {
  int i = blockIdx.x * 256 + threadIdx.x;
  if (i >= NN) return;
  const float* e = emb + (size_t)i * 64;
  float s = 0.f;
#pragma unroll
  for (int d = 0; d < 64; ++d) s = fmaf(e[d], e[d], s);
  sq[i] = s;
}

// ---------------------------------------------------------------------------
// dist + top-15 via WMMA.
// Block = 16 rows, 8 waves. Per 128-col chunk: wave w computes the 16x16
// dot-product tile at cols [chunk + 16w) with 16 steps of wmma_f32_16x16x4,
// converts to distances in LDS, then each half-wave team merges one row's
// 128 candidates into that row's top-15 pool (replace-max, shuffle reduce).
// The top-15 is kept UNSORTED: the reference only needs the edge set and
// eps = max of the 15 (computed by eps_kernel below).
// ---------------------------------------------------------------------------
__global__ __launch_bounds__(256) void dist_topk_kernel(
    const float* __restrict__ emb, const float* __restrict__ sq,
    int* __restrict__ knn_idx, float* __restrict__ knn_dist) {
  __shared__ float sEmb[16 * 64];
  __shared__ float sSq[16];
  __shared__ float sDist[16][CHUNK];
  __shared__ float sPoolV[16][16];
  __shared__ int   sPoolI[16][16];

  const int row0 = blockIdx.x * 16;
  const int tid  = threadIdx.x;
  const int wave = tid >> 5;
  const int lane = tid & 31;
  const int ln   = lane & 15;
  const int koff = (lane < 16) ? 0 : 2;
  const int moff = (lane < 16) ? 0 : 8;

  for (int t = tid; t < 16 * 64; t += 256) sEmb[t] = emb[(size_t)row0 * 64 + t];
  if (tid < 16) sSq[tid] = sq[row0 + tid];
  sPoolV[tid >> 4][tid & 15] = __builtin_inff();
  sPoolI[tid >> 4][tid & 15] = -1;
  __syncthreads();

  // per-team (half-wave) replicated state
  const int mrow = 2 * wave + (lane >> 4);  // row merged by this 16-lane team
  const int tl   = ln;
  float poolMax = __builtin_inff();
  int   maxSlot = 0;

  for (int cb = 0; cb < NN; cb += CHUNK) {
    // ---- WMMA tile: dot(row m, col) for cols [cb + wave*16, +16)
    const int col = cb + wave * 16 + ln;
    v8f c = {0.f, 0.f, 0.f, 0.f, 0.f, 0.f, 0.f, 0.f};
    const float* ecol = emb + (size_t)col * 64;
#pragma unroll
    for (int kk = 0; kk < 16; ++kk) {
      int k0 = 4 * kk + koff;
      v2f a  = {sEmb[ln * 64 + k0], sEmb[ln * 64 + k0 + 1]};
      v2f bf = {ecol[k0], ecol[k0 + 1]};
      c = __builtin_amdgcn_wmma_f32_16x16x4_f32(false, a, false, bf,
                                                (short)0, c, false, false);
    }
    const float sqc = sq[col];
#pragma unroll
    for (int r = 0; r < 8; ++r) {
      int m = r + moff;
      float d2 = sSq[m] + sqc - 2.0f * c[r];
      float ds = sqrtf(fmaxf(d2, 0.0f));
      sDist[m][wave * 16 + ln] = (col == row0 + m) ? __builtin_inff() : ds;
    }
    __syncthreads();

    // ---- merge chunk into per-row pool (16-lane team, no block barriers)
    for (;;) {
      float v = __builtin_inff();
      int   ci = 0x7fffffff;
#pragma unroll
      for (int j = 0; j < CHUNK / 16; ++j) {
        float x = sDist[mrow][tl + 16 * j];
        int   xc = cb + tl + 16 * j;
        if (x < v || (x == v && xc < ci)) { v = x; ci = xc; }
      }
#pragma unroll
      for (int off = 8; off >= 1; off >>= 1) {
        float ov = __shfl_xor(v, off, 16);
        int   oi = __shfl_xor(ci, off, 16);
        if (ov < v || (ov == v && oi < ci)) { v = ov; ci = oi; }
      }
      if (!(v < poolMax)) break;           // consistent across the team
      if (tl == 0) {
        sPoolV[mrow][maxSlot] = v;
        sPoolI[mrow][maxSlot] = ci;
        sDist[mrow][ci - cb]  = __builtin_inff();   // consume candidate
      }
      // recompute pool max (replicated; same-wave LDS ops are in order)
      float pv = (tl < KNEI) ? sPoolV[mrow][tl] : -1.0f;
      int   ps = tl;
#pragma unroll
      for (int off = 8; off >= 1; off >>= 1) {
        float ov = __shfl_xor(pv, off, 16);
        int   oi = __shfl_xor(ps, off, 16);
        if (ov > pv) { pv = ov; ps = oi; }
      }
      poolMax = pv;
      maxSlot = ps;
    }
    __syncthreads();
  }

  // write out pools (unsorted top-15)
  if (tid < 16 * KNEI) {
    int r = tid / KNEI, k2 = tid % KNEI;
    knn_dist[(size_t)(row0 + r) * KNEI + k2] = sPoolV[r][k2];
    knn_idx [(size_t)(row0 + r) * KNEI + k2] = sPoolI[r][k2];
  }
}

// eps[i] = max over the row's 15 neighbor distances (adaptive bandwidth)
__global__ __launch_bounds__(256) void eps_kernel(const float* __restrict__ knn_dist,
                                                  float* __restrict__ eps) {
  int i = blockIdx.x * 256 + threadIdx.x;
  if (i >= NN) return;
  float m = 0.f;
#pragma unroll
  for (int k = 0; k < KNEI; ++k) m = fmaxf(m, knn_dist[i * KNEI + k]);
  eps[i] = m;
}

// ---------------------------------------------------------------------------
// adaptive-bandwidth gaussian weights + threshold
// ---------------------------------------------------------------------------
__global__ __launch_bounds__(256) void weight_kernel(
    const float* __restrict__ knn_dist, const int* __restrict__ knn_idx,
    const float* __restrict__ eps, float* __restrict__ wbuf) {
  int e = blockIdx.x * 256 + threadIdx.x;
  if (e >= M_NEI) return;
  int i = e / KNEI;
  float nd   = knn_dist[e];
  float epsi = eps[i];
  float epsj = eps[knn_idx[e]];
  float w = 0.5f * __expf(-__powf(nd / epsi, 0.95f)) +
            0.5f * __expf(-__powf(nd / epsj, 0.95f));
  wbuf[e] = (w > 1.0e-4f) ? w : 0.0f;
}

// deg[col] += w  (deg pre-initialized to 1.0 for self loop)
__global__ __launch_bounds__(256) void deg_kernel(const int* __restrict__ knn_idx,
                                                  const float* __restrict__ wbuf,
                                                  float* __restrict__ deg) {
  int e = blockIdx.x * 256 + threadIdx.x;
  if (e >= M_NEI) return;
  atomicAdd(&deg[knn_idx[e]], wbuf[e]);
}

__global__ __launch_bounds__(256) void dis_kernel(const float* __restrict__ deg,
                                                  float* __restrict__ dis) {
  int i = blockIdx.x * 256 + threadIdx.x;
  if (i >= NN) return;
  float d = deg[i];
  dis[i] = (d > 0.0f) ? rsqrtf(d) : 0.0f;
}

// count incoming edges per destination (in_cnt pre-initialized to 1 for self)
__global__ __launch_bounds__(256) void cnt_kernel(const int* __restrict__ knn_idx,
                                                  int* __restrict__ in_cnt) {
  int e = blockIdx.x * 256 + threadIdx.x;
  if (e >= M_NEI) return;
  atomicAdd(&in_cnt[knn_idx[e]], 1);
}

// single-block exclusive scan over 8192 counts -> offsets (+ cursor copy)
__global__ __launch_bounds__(256) void scan_kernel(const int* __restrict__ cnt,
                                                   int* __restrict__ offs,
                                                   int* __restrict__ cursor) {
  __shared__ int part[256];
  __shared__ int pref[257];
  const int t = threadIdx.x;
  const int base = t * 32;
  int s = 0;
  for (int j = 0; j < 32; ++j) s += cnt[base + j];
  part[t] = s;
  __syncthreads();
  if (t == 0) {
    int acc = 0;
    for (int j = 0; j < 256; ++j) { pref[j] = acc; acc += part[j]; }
    pref[256] = acc;
  }
  __syncthreads();
  int acc = pref[t];
  for (int j = 0; j < 32; ++j) {
    offs[base + j] = acc;
    cursor[base + j] = acc;
    acc += cnt[base + j];
  }
  if (t == 255) offs[NN] = pref[256];
}

// scatter edges into CSC with per-edge gcn norm; self loops at tail
__global__ __launch_bounds__(256) void fill_kernel(
    const int* __restrict__ knn_idx, const float* __restrict__ wbuf,
    const float* __restrict__ dis, int* __restrict__ cursor,
    int* __restrict__ csc_row, float* __restrict__ csc_norm) {
  int e = blockIdx.x * 256 + threadIdx.x;
  if (e >= M_TOT) return;
  if (e < M_NEI) {
    int r = e / KNEI;
    int c = knn_idx[e];
    float nrm = dis[r] * wbuf[e] * dis[c];
    int p = atomicAdd(&cursor[c], 1);
    csc_row[p]  = r;
    csc_norm[p] = nrm;
  } else {
    int i = e - M_NEI;
    int p = atomicAdd(&cursor[i], 1);
    csc_row[p]  = i;
    csc_norm[p] = dis[i] * dis[i];
  }
}

// ---------------------------------------------------------------------------
// MLP: per node, X (16x128) -> relu(X@W1^T+b1) -> (@W2^T+b2), fp32 WMMA.
// One wave owns one 16-wide output tile; 8 waves/block cover H=D=128.
// Writes x (node-major) and hidden = temp[0]*x.
// ---------------------------------------------------------------------------
#define NPB 4
__global__ __launch_bounds__(256) void mlp_kernel(
    const float* __restrict__ gdb,          // (B, N, D)
    const float* __restrict__ W1, const float* __restrict__ b1,
    const float* __restrict__ W2, const float* __restrict__ b2,
    const float* __restrict__ temp,
    float* __restrict__ xout, float* __restrict__ hidden) {
  __shared__ float sX[BB * DD];
  __shared__ float sH[BB * HH];
  const int wave = threadIdx.x >> 5;
  const int lane = threadIdx.x & 31;
  const int ln   = lane & 15;
  const int koff = (lane < 16) ? 0 : 2;
  const int moff = (lane < 16) ? 0 : 8;
  const float t0 = temp[0];
  const int node0 = blockIdx.x * NPB;
  for (int node = node0; node < node0 + NPB; ++node) {
    __syncthreads();
    for (int t = threadIdx.x; t < BB * DD; t += 256) {
      int b = t >> 7, d = t & 127;
      sX[t] = gdb[((size_t)b * NN + node) * DD + d];
    }
    __syncthreads();
    v8f c1 = {0.f, 0.f, 0.f, 0.f, 0.f, 0.f, 0.f, 0.f};
    const float* w1row = W1 + (size_t)(wave * 16 + ln) * DD;
#pragma unroll 4
    for (int kk = 0; kk < 32; ++kk) {
      int k0 = 4 * kk + koff;
      v2f a  = {sX[ln * DD + k0], sX[ln * DD + k0 + 1]};
      v2f bf = {w1row[k0], w1row[k0 + 1]};
      c1 = __builtin_amdgcn_wmma_f32_16x16x4_f32(false, a, false, bf,
                                                 (short)0, c1, false, false);
    }
    float bias1 = b1[wave * 16 + ln];
#pragma unroll
    for (int r = 0; r < 8; ++r) {
      float v = c1[r] + bias1;
      sH[(r + moff) * HH + wave * 16 + ln] = (v > 0.f) ? v : 0.f;
    }
    __syncthreads();
    v8f c2 = {0.f, 0.f, 0.f, 0.f, 0.f, 0.f, 0.f, 0.f};
    const float* w2row = W2 + (size_t)(wave * 16 + ln) * HH;
#pragma unroll 4
    for (int kk = 0; kk < 32; ++kk) {
      int k0 = 4 * kk + koff;
      v2f a  = {sH[ln * HH + k0], sH[ln * HH + k0 + 1]};
      v2f bf = {w2row[k0], w2row[k0 + 1]};
      c2 = __builtin_amdgcn_wmma_f32_16x16x4_f32(false, a, false, bf,
                                                 (short)0, c2, false, false);
    }
    float bias2 = b2[wave * 16 + ln];
#pragma unroll
    for (int r = 0; r < 8; ++r) {
      float v = c2[r] + bias2;
      size_t o = (size_t)node * BD + (size_t)(r + moff) * DD + wave * 16 + ln;
      xout[o]   = v;
      hidden[o] = t0 * v;
    }
  }
}

// ---------------------------------------------------------------------------
// one GPR-GNN hop: out[c,:] = sum_{(r,nrm) in CSC[c]} nrm * xin[r,:]
// hidden[c,:] += temp[k] * out[c,:].  One block per destination node.
// Neighbor feature rows (8KB) are double-buffered into LDS with
// global_load_async_to_lds_b128; each lane stages and consumes its own 32B,
// so no barriers are needed, only s_wait_asynccnt.
// ---------------------------------------------------------------------------
__global__ __launch_bounds__(256) void hop_kernel(
    const float* __restrict__ xin, float* __restrict__ xout,
    float* __restrict__ hidden, const int* __restrict__ offs,
    const int* __restrict__ csc_row, const float* __restrict__ csc_norm,
    const float* __restrict__ temp, int k) {
  __shared__ float stage[2][BD];
  const int c   = blockIdx.x;
  const int tid = threadIdx.x;
  const float gamma = temp[k];
  const int beg = offs[c], end = offs[c + 1];

  float4 a0 = {0.f, 0.f, 0.f, 0.f}, a1 = {0.f, 0.f, 0.f, 0.f};

  {  // stage first edge row into buffer 0
    const float* src = xin + (size_t)csc_row[beg] * BD + tid * 8;
    unsigned lds = (unsigned)(size_t)&stage[0][tid * 8];
    asm volatile("global_load_async_to_lds_b128 %0, %1, off\n\t"
                 "global_load_async_to_lds_b128 %0, %1, off offset:16"
                 :: "v"(lds), "v"(src) : "memory");
  }
  for (int e = beg; e < end; ++e) {
    const int cur = (e - beg) & 1;
    if (e + 1 < end) {
      const float* src = xin + (size_t)csc_row[e + 1] * BD + tid * 8;
      unsigned lds = (unsigned)(size_t)&stage[cur ^ 1][tid * 8];
      asm volatile("global_load_async_to_lds_b128 %0, %1, off\n\t"
                   "global_load_async_to_lds_b128 %0, %1, off offset:16"
                   :: "v"(lds), "v"(src) : "memory");
      asm volatile("s_wait_asynccnt 0x2" ::: "memory");  // current stage done
    } else {
      asm volatile("s_wait_asynccnt 0x0" ::: "memory");
    }
    float nrm = csc_norm[e];
    const float4* xr = (const float4*)&stage[cur][tid * 8];
    float4 p0 = xr[0], p1 = xr[1];
    a0.x = fmaf(nrm, p0.x, a0.x); a0.y = fmaf(nrm, p0.y, a0.y);
    a0.z = fmaf(nrm, p0.z, a0.z); a0.w = fmaf(nrm, p0.w, a0.w);
    a1.x = fmaf(nrm, p1.x, a1.x); a1.y = fmaf(nrm, p1.y, a1.y);
    a1.z = fmaf(nrm, p1.z, a1.z); a1.w = fmaf(nrm, p1.w, a1.w);
  }
  float4* xo = (float4*)(xout + (size_t)c * BD + tid * 8);
  float4* hp = (float4*)(hidden + (size_t)c * BD + tid * 8);
  xo[0] = a0; xo[1] = a1;
  float4 h0 = hp[0], h1 = hp[1];
  h0.x += gamma * a0.x; h0.y += gamma * a0.y;
  h0.z += gamma * a0.z; h0.w += gamma * a0.w;
  h1.x += gamma * a1.x; h1.y += gamma * a1.y;
  h1.z += gamma * a1.z; h1.w += gamma * a1.w;
  hp[0] = h0; hp[1] = h1;
}

// ---------------------------------------------------------------------------
// gene_rep = mean over nodes of hidden  (two-phase: atomic partials + scale)
// ---------------------------------------------------------------------------
__global__ __launch_bounds__(256) void mean_kernel(const float* __restrict__ hidden,
                                                   float* __restrict__ out) {
  const int o    = (blockIdx.x & 7) * 256 + threadIdx.x;   // 0..2047
  const int nbeg = (blockIdx.x >> 3) * 1024;               // 8 node-chunks
  float s = 0.f;
  for (int n = nbeg; n < nbeg + 1024; ++n) s += hidden[(size_t)n * BD + o];
  atomicAdd(&out[o], s);
}

__global__ __launch_bounds__(256) void scale_kernel(float* __restrict__ out) {
  int o = blockIdx.x * 256 + threadIdx.x;
  if (o < BD) out[o] *= (1.0f / (float)NN);
}

// ---------------------------------------------------------------------------
// pred = gene_rep @ Wp^T + bp   (16 x 8192, K=128) via fp32 WMMA
// ---------------------------------------------------------------------------
__global__ __launch_bounds__(256) void pred_kernel(
    const float* __restrict__ gene_rep, const float* __restrict__ Wp,
    const float* __restrict__ bp, float* __restrict__ pred) {
  __shared__ float sA[BB * DD];
  for (int t = threadIdx.x; t < BB * DD; t += 256) sA[t] = gene_rep[t];
  __syncthreads();
  const int wave = threadIdx.x >> 5;
  const int lane = threadIdx.x & 31;
  const int ln   = lane & 15;
  const int koff = (lane < 16) ? 0 : 2;
  const int moff = (lane < 16) ? 0 : 8;
  const int tile = blockIdx.x * 8 + wave;   // 0..511
  v8f c = {0.f, 0.f, 0.f, 0.f, 0.f, 0.f, 0.f, 0.f};
  const float* wrow = Wp + (size_t)(tile * 16 + ln) * DD;
#pragma unroll 4
  for (int kk = 0; kk < 32; ++kk) {
    int k0 = 4 * kk + koff;
    v2f a  = {sA[ln * DD + k0], sA[ln * DD + k0 + 1]};
    v2f bf = {wrow[k0], wrow[k0 + 1]};
    c = __builtin_amdgcn_wmma_f32_16x16x4_f32(false, a, false, bf,
                                              (short)0, c, false, false);
  }
  float bias = bp[tile * 16 + ln];
#pragma unroll
  for (int r = 0; r < 8; ++r)
    pred[(size_t)(r + moff) * NN + tile * 16 + ln] = c[r] + bias;
}

// ---------------------------------------------------------------------------
extern "C" void kernel_launch(void* const* d_in, const int* in_sizes, int n_in,
                              void* d_out, int out_size, void* d_ws, size_t ws_size,
                              hipStream_t stream) {
  const float* gdb  = (const float*)d_in[0];  // (16, 8192, 128)
  const float* cemb = (const float*)d_in[1];  // (8192, 64)
  // d_in[2] gene_indices: unused by the reference computation
  const float* W1   = (const float*)d_in[3];
  const float* b1   = (const float*)d_in[4];
  const float* W2   = (const float*)d_in[5];
  const float* b2   = (const float*)d_in[6];
  const float* temp = (const float*)d_in[7];  // 11 gammas
  const float* Wp   = (const float*)d_in[8];
  const float* bp   = (const float*)d_in[9];

  float* out_rep  = (float*)d_out;            // 2048
  float* out_pred = (float*)d_out + BD;       // 16*8192

  // workspace partition (~203 MB)
  float* x0       = (float*)d_ws;
  float* x1       = x0 + (size_t)NN * BD;
  float* hidden   = x1 + (size_t)NN * BD;
  float* sq       = hidden + (size_t)NN * BD;
  float* knn_dist = sq + NN;
  float* wbuf     = knn_dist + M_NEI;
  float* deg      = wbuf + M_NEI;
  float* dis      = deg + NN;
  float* eps      = dis + NN;
  float* csc_norm = eps + NN;
  int*   knn_idx  = (int*)(csc_norm + M_TOT);
  int*   in_cnt   = knn_idx + M_NEI;
  int*   offs     = in_cnt + NN;
  int*   cursor   = offs + (NN + 1);
  int*   csc_row  = cursor + NN;

  // ---- graph construction
  init_kernel<<<32, 256, 0, stream>>>(deg, in_cnt, out_rep);
  sqnorm_kernel<<<32, 256, 0, stream>>>(cemb, sq);
  dist_topk_kernel<<<NN / 16, 256, 0, stream>>>(cemb, sq, knn_idx, knn_dist);
  eps_kernel<<<32, 256, 0, stream>>>(knn_dist, eps);
  weight_kernel<<<(M_NEI + 255) / 256, 256, 0, stream>>>(knn_dist, knn_idx, eps, wbuf);
  deg_kernel<<<(M_NEI + 255) / 256, 256, 0, stream>>>(knn_idx, wbuf, deg);
  dis_kernel<<<32, 256, 0, stream>>>(deg, dis);
  cnt_kernel<<<(M_NEI + 255) / 256, 256, 0, stream>>>(knn_idx, in_cnt);
  scan_kernel<<<1, 256, 0, stream>>>(in_cnt, offs, cursor);
  fill_kernel<<<(M_TOT + 255) / 256, 256, 0, stream>>>(knn_idx, wbuf, dis, cursor,
                                                       csc_row, csc_norm);
  // ---- MLP (WMMA) producing x0 and hidden = temp[0]*x0
  mlp_kernel<<<NN / NPB, 256, 0, stream>>>(gdb, W1, b1, W2, b2, temp, x0, hidden);

  // ---- 10 propagation hops (ping-pong, L2-resident)
  float* xin = x0;
  float* xo  = x1;
  for (int k = 1; k <= 10; ++k) {
    hop_kernel<<<NN, 256, 0, stream>>>(xin, xo, hidden, offs, csc_row, csc_norm,
                                       temp, k);
    float* t = xin; xin = xo; xo = t;
  }

  // ---- readout
  mean_kernel<<<64, 256, 0, stream>>>(hidden, out_rep);
  scale_kernel<<<8, 256, 0, stream>>>(out_rep);
  pred_kernel<<<64, 256, 0, stream>>>(out_rep, Wp, bp, out_pred);

  (void)in_sizes; (void)n_in; (void)out_size; (void)ws_size;
}